// warp_67388036874445
// MI455X (gfx1250) — compile-verified
//
#include <hip/hip_runtime.h>
#include <cmath>

#ifndef __has_builtin
#define __has_builtin(x) 0
#endif

// Problem dimensions (from the reference)
constexpr int Bn = 2, Cn = 64, Hn = 512, Wn = 896;
constexpr int HW = Hn * Wn;               // 458752 (divisible by 256)
constexpr int NPIX = Bn * Hn * Wn;        // 917504 pixels
#define TPB 256

#if defined(__HIP_DEVICE_COMPILE__)
#define HAS_ASYNC_LDS \
  (__has_builtin(__builtin_amdgcn_global_load_async_to_lds_b32) && \
   __has_builtin(__builtin_amdgcn_s_wait_asynccnt))
#else
#define HAS_ASYNC_LDS 0
#endif

__global__ __launch_bounds__(TPB) void resample2d_kernel(
    const float* __restrict__ feature,   // [B,C,H,W]
    const float* __restrict__ flow,      // [B,2,H,W]
    float* __restrict__ out)             // [B,C,H,W]
{
    __shared__ float s_u[TPB];
    __shared__ float s_v[TPB];

    const int tid = threadIdx.x;
    const int pix = blockIdx.x * TPB + tid;
    if (pix >= NPIX) return;

    const int b = pix / HW;
    const int p = pix - b * HW;          // y*W + x within the plane
    const int y = p / Wn;
    const int x = p - y * Wn;

    // ---- stage flow (u,v) for this pixel through LDS via async copy ----
    const float* gu = flow + (size_t)b * 2 * HW + p;   // u plane
    const float* gv = gu + HW;                          // v plane

#if HAS_ASYNC_LDS
    typedef __attribute__((address_space(1))) int* g32_t;   // global int*
    typedef __attribute__((address_space(3))) int* l32_t;   // LDS int*
    __builtin_amdgcn_global_load_async_to_lds_b32(
        (g32_t)(float*)gu, (l32_t)&s_u[tid], 0, 0);
    __builtin_amdgcn_global_load_async_to_lds_b32(
        (g32_t)(float*)gv, (l32_t)&s_v[tid], 0, 0);
    __builtin_amdgcn_s_wait_asynccnt(0);
    const float u = s_u[tid];
    const float v = s_v[tid];
#else
    const float u = *gu;
    const float v = *gv;
#endif

    // ---- bilinear setup (shared across all 64 channels) ----
    const float gx = (float)x + u;
    const float gy = (float)y + v;
    const float x0f = floorf(gx);
    const float y0f = floorf(gy);
    const float wx = gx - x0f;
    const float wy = gy - y0f;
    const int x0 = (int)x0f;
    const int y0 = (int)y0f;
    const int x1 = x0 + 1;
    const int y1 = y0 + 1;

    const bool vx0 = (x0 >= 0) && (x0 < Wn);
    const bool vx1 = (x1 >= 0) && (x1 < Wn);
    const bool vy0 = (y0 >= 0) && (y0 < Hn);
    const bool vy1 = (y1 >= 0) && (y1 < Hn);

    const int cx0 = x0 < 0 ? 0 : (x0 > Wn - 1 ? Wn - 1 : x0);
    const int cx1 = x1 < 0 ? 0 : (x1 > Wn - 1 ? Wn - 1 : x1);
    const int cy0 = y0 < 0 ? 0 : (y0 > Hn - 1 ? Hn - 1 : y0);
    const int cy1 = y1 < 0 ? 0 : (y1 > Hn - 1 ? Hn - 1 : y1);

    const int off00 = cy0 * Wn + cx0;
    const int off01 = cy0 * Wn + cx1;
    const int off10 = cy1 * Wn + cx0;
    const int off11 = cy1 * Wn + cx1;

    const float w00 = (vx0 && vy0) ? (1.0f - wx) * (1.0f - wy) : 0.0f;
    const float w01 = (vx1 && vy0) ? wx * (1.0f - wy) : 0.0f;
    const float w10 = (vx0 && vy1) ? (1.0f - wx) * wy : 0.0f;
    const float w11 = (vx1 && vy1) ? wx * wy : 0.0f;

    const float* f0 = feature + (size_t)b * Cn * HW;
    float* o = out + (size_t)b * Cn * HW + p;

    // Warm the first tap cachelines (global_prefetch_b8 on gfx1250)
    __builtin_prefetch(f0 + off00, 0, 3);
    __builtin_prefetch(f0 + off10, 0, 3);

    // ---- sweep channels: same taps/weights, plane stride HW ----
#pragma unroll 8
    for (int c = 0; c < Cn; ++c) {
        const float* pl = f0 + (size_t)c * HW;
        float acc = w00 * pl[off00];
        acc = fmaf(w01, pl[off01], acc);
        acc = fmaf(w10, pl[off10], acc);
        acc = fmaf(w11, pl[off11], acc);
        o[(size_t)c * HW] = acc;
    }
}

extern "C" void kernel_launch(void* const* d_in, const int* in_sizes, int n_in,
                              void* d_out, int out_size, void* d_ws, size_t ws_size,
                              hipStream_t stream) {
    const float* feature = (const float*)d_in[0];   // [B,C,H,W] f32
    const float* flow    = (const float*)d_in[1];   // [B,2,H,W] f32
    float* outp          = (float*)d_out;           // [B,C,H,W] f32

    const int blocks = (NPIX + TPB - 1) / TPB;      // 3584 blocks of 256 (8 waves)
    resample2d_kernel<<<blocks, TPB, 0, stream>>>(feature, flow, outp);
}